// CrossViewAttention_13692355740441
// MI455X (gfx1250) — compile-verified
//
#include <hip/hip_runtime.h>
#include <math.h>

typedef __attribute__((ext_vector_type(2))) float v2f;
typedef __attribute__((ext_vector_type(8))) float v8f;

#define DIMC 128
#define NCAM 6
#define KK   1680   // 28*60 keys per camera
#define QQ   2500   // 50*50 queries
#define QP   2512   // QQ padded to 157*16
#define HEADS_C 4
#define DH_C 32
#define ATT_SCALE 0.17677669529663687f  // 32^-0.5

// ---------------------------------------------------------------------------
// fp32 WMMA helper: D = A(16x4) * B(4x16) + C  (V_WMMA_F32_16X16X4_F32)
// A layout: lanes 0-15 M=0..15 {v0:K0,v1:K1}; lanes 16-31 M=0..15 {v0:K2,v1:K3}
// B layout (mirror): lanes 0-15 N=0..15 {v0:K0,v1:K1}; lanes16-31 {v0:K2,v1:K3}
// C layout: vgpr r: lanes0-15 -> (M=r,N=lane); lanes16-31 -> (M=r+8,N=lane-16)
// ---------------------------------------------------------------------------
__device__ __forceinline__ v8f wmma4(v2f a, v2f b, v8f c) {
  return __builtin_amdgcn_wmma_f32_16x16x4_f32(false, a, false, b, (short)0, c, false, false);
}

// ------------------------- geometry / embedding kernels ---------------------
__global__ void k_cembed(const float* __restrict__ E_inv, const float* __restrict__ w_cam,
                         float* __restrict__ c_embed) {
  int i = blockIdx.x * blockDim.x + threadIdx.x;
  if (i >= NCAM * DIMC) return;
  int n = i / DIMC, d = i % DIMC;
  const float* E = E_inv + n * 16;
  float s = 0.f;
#pragma unroll
  for (int j = 0; j < 4; ++j) s += E[j * 4 + 3] * w_cam[d * 4 + j];
  c_embed[i] = s;
}

__global__ void k_imgembed(const float* __restrict__ I_inv, const float* __restrict__ E_inv,
                           const float* __restrict__ image_plane, const float* __restrict__ w_img,
                           const float* __restrict__ c_embed, float* __restrict__ keyin) {
  int i = blockIdx.x * blockDim.x + threadIdx.x;
  if (i >= NCAM * KK) return;
  int n = i / KK, k = i % KK;
  float p0 = image_plane[k], p1 = image_plane[KK + k], p2 = image_plane[2 * KK + k];
  const float* I = I_inv + n * 9;
  float cam[4];
  cam[0] = I[0] * p0 + I[1] * p1 + I[2] * p2;
  cam[1] = I[3] * p0 + I[4] * p1 + I[5] * p2;
  cam[2] = I[6] * p0 + I[7] * p1 + I[8] * p2;
  cam[3] = 1.f;
  const float* E = E_inv + n * 16;
  float dv[4];
#pragma unroll
  for (int r = 0; r < 4; ++r)
    dv[r] = E[r*4+0]*cam[0] + E[r*4+1]*cam[1] + E[r*4+2]*cam[2] + E[r*4+3]*cam[3];
  const float* ce = c_embed + n * DIMC;
  float nrm = 0.f;
  for (int d = 0; d < DIMC; ++d) {
    float t = w_img[d*4+0]*dv[0] + w_img[d*4+1]*dv[1] + w_img[d*4+2]*dv[2] + w_img[d*4+3]*dv[3] - ce[d];
    nrm += t * t;
  }
  float inv = 1.f / (sqrtf(nrm) + 1e-7f);
  float* out = keyin + (size_t)(n * KK + k) * DIMC;
  for (int d = 0; d < DIMC; ++d) {
    float t = w_img[d*4+0]*dv[0] + w_img[d*4+1]*dv[1] + w_img[d*4+2]*dv[2] + w_img[d*4+3]*dv[3] - ce[d];
    out[d] = t * inv;
  }
}

__global__ void k_queryin(const float* __restrict__ world, const float* __restrict__ w_bev,
                          const float* __restrict__ b_bev, const float* __restrict__ c_embed,
                          const float* __restrict__ x, float* __restrict__ queryin) {
  int i = blockIdx.x * blockDim.x + threadIdx.x;
  if (i >= NCAM * QQ) return;
  int n = i / QQ, q = i % QQ;
  float w0 = world[q], w1 = world[QQ + q];
  const float* ce = c_embed + n * DIMC;
  float nrm = 0.f;
  for (int d = 0; d < DIMC; ++d) {
    float t = w_bev[d*2] * w0 + w_bev[d*2+1] * w1 + b_bev[d] - ce[d];
    nrm += t * t;
  }
  float inv = 1.f / (sqrtf(nrm) + 1e-7f);
  float* out = queryin + (size_t)(n * QP + q) * DIMC;
  for (int d = 0; d < DIMC; ++d) {
    float t = w_bev[d*2] * w0 + w_bev[d*2+1] * w1 + b_bev[d] - ce[d];
    out[d] = t * inv + x[(size_t)d * QQ + q];
  }
}

// BN+ReLU, writing activations TRANSPOSED: act_t[(n*KK + k)*DIMC + c]
__global__ void k_bnrelu(const float* __restrict__ feat,
                         const float* __restrict__ pg, const float* __restrict__ pb,
                         const float* __restrict__ pm, const float* __restrict__ pv,
                         const float* __restrict__ lg, const float* __restrict__ lb,
                         const float* __restrict__ lm, const float* __restrict__ lv,
                         float* __restrict__ actp_t, float* __restrict__ actl_t) {
  size_t i = (size_t)blockIdx.x * blockDim.x + threadIdx.x;
  if (i >= (size_t)NCAM * DIMC * KK) return;
  int k = (int)(i % KK);
  int c = (int)((i / KK) % DIMC);
  int n = (int)(i / ((size_t)DIMC * KK));
  float f = feat[i];
  size_t o = (size_t)(n * KK + k) * DIMC + c;
  float sp = pg[c] * rsqrtf(pv[c] + 1e-5f);
  float vp = (f - pm[c]) * sp + pb[c];
  actp_t[o] = vp > 0.f ? vp : 0.f;
  float sl = lg[c] * rsqrtf(lv[c] + 1e-5f);
  float vl = (f - lm[c]) * sl + lb[c];
  actl_t[o] = vl > 0.f ? vl : 0.f;
}

// --------------------------- fp32 WMMA GEMM ---------------------------------
// One wave computes a 16x64 tile of C = A(MxKr) * B^T (+Cin) (+bias).
// A row-major: A[m*Kr + kc];  B is a weight (N, Kr): B_el(kc, nn) = B[nn*Kr+kc]
// Software-pipelined: next-step fragments load while current WMMAs execute.
__global__ __launch_bounds__(32)
void k_gemm(const float* __restrict__ A, int Kr,
            const float* __restrict__ B,
            const float* __restrict__ Cin, const float* __restrict__ bias,
            float* __restrict__ Cout, int N) {
  int l = threadIdx.x;
  int hf = l >> 4, lmn = l & 15;
  int m0 = blockIdx.x * 16, n0 = blockIdx.y * 64;
  v8f c0, c1, c2, c3;
#pragma unroll
  for (int r = 0; r < 8; ++r) {
    int row = m0 + r + 8 * hf;
    float v0 = 0.f, v1 = 0.f, v2 = 0.f, v3 = 0.f;
    if (Cin) {
      const float* cp = Cin + (size_t)row * N + n0 + lmn;
      v0 = cp[0]; v1 = cp[16]; v2 = cp[32]; v3 = cp[48];
    }
    if (bias) {
      const float* bp = bias + n0 + lmn;
      v0 += bp[0]; v1 += bp[16]; v2 += bp[32]; v3 += bp[48];
    }
    c0[r] = v0; c1[r] = v1; c2[r] = v2; c3[r] = v3;
  }
  const float* Ap = A + (size_t)(m0 + lmn) * Kr + 2 * hf;
  const float* Bp = B + (size_t)(n0 + lmn) * Kr + 2 * hf;
  v2f a0 = *(const v2f*)(Ap);               // 8B-aligned (Kr even, 2*hf even)
  v2f b0 = *(const v2f*)(Bp);
  v2f b1 = *(const v2f*)(Bp + (size_t)16 * Kr);
  v2f b2 = *(const v2f*)(Bp + (size_t)32 * Kr);
  v2f b3 = *(const v2f*)(Bp + (size_t)48 * Kr);
  for (int kc = 4; kc < Kr; kc += 4) {
    v2f an  = *(const v2f*)(Ap + kc);
    v2f bn0 = *(const v2f*)(Bp + kc);
    v2f bn1 = *(const v2f*)(Bp + (size_t)16 * Kr + kc);
    v2f bn2 = *(const v2f*)(Bp + (size_t)32 * Kr + kc);
    v2f bn3 = *(const v2f*)(Bp + (size_t)48 * Kr + kc);
    c0 = wmma4(a0, b0, c0);
    c1 = wmma4(a0, b1, c1);
    c2 = wmma4(a0, b2, c2);
    c3 = wmma4(a0, b3, c3);
    a0 = an; b0 = bn0; b1 = bn1; b2 = bn2; b3 = bn3;
  }
  c0 = wmma4(a0, b0, c0);
  c1 = wmma4(a0, b1, c1);
  c2 = wmma4(a0, b2, c2);
  c3 = wmma4(a0, b3, c3);
#pragma unroll
  for (int r = 0; r < 8; ++r) {
    float* op = Cout + (size_t)(m0 + r + 8 * hf) * N + n0 + lmn;
    op[0] = c0[r]; op[16] = c1[r]; op[32] = c2[r]; op[48] = c3[r];
  }
}

// ------------------------------- LayerNorm ----------------------------------
// rmode: 0 none, 1 resid same layout [row*128+d], 2 resid transposed [d*Qres+row]
__global__ void k_ln(const float* __restrict__ in, const float* __restrict__ resid,
                     int rmode, int Qres,
                     const float* __restrict__ g, const float* __restrict__ bb,
                     float* __restrict__ out, long out_sr, long out_sd, int rows) {
  int gid = blockIdx.x * blockDim.x + threadIdx.x;
  int row = gid >> 5, lane = gid & 31;
  if (row >= rows) return;
  const float* ip = in + (size_t)row * DIMC;
  float v[4];
  float s = 0.f;
#pragma unroll
  for (int i = 0; i < 4; ++i) {
    int d = lane + 32 * i;
    float t = ip[d];
    if (rmode == 1) t += resid[(size_t)row * DIMC + d];
    else if (rmode == 2) t += resid[(size_t)d * Qres + row];
    v[i] = t; s += t;
  }
#pragma unroll
  for (int o = 16; o > 0; o >>= 1) s += __shfl_xor(s, o, 32);
  float mu = s * (1.f / DIMC);
  float var = 0.f;
#pragma unroll
  for (int i = 0; i < 4; ++i) { float dl = v[i] - mu; var += dl * dl; }
#pragma unroll
  for (int o = 16; o > 0; o >>= 1) var += __shfl_xor(var, o, 32);
  float sc = rsqrtf(var * (1.f / DIMC) + 1e-5f);
#pragma unroll
  for (int i = 0; i < 4; ++i) {
    int d = lane + 32 * i;
    out[(size_t)row * out_sr + (size_t)d * out_sd] = (v[i] - mu) * sc * g[d] + bb[d];
  }
}

__global__ void k_gelu(float* __restrict__ h, int n) {
  int i = blockIdx.x * blockDim.x + threadIdx.x;
  if (i < n) {
    float x = h[i];
    h[i] = 0.5f * x * (1.f + erff(x * 0.70710678118654752f));
  }
}

// --------------- attention pass 1 (per camera): row max / sum-exp -----------
__global__ __launch_bounds__(32)
void k_attn_pass1(const float* __restrict__ qbuf, const float* __restrict__ kbuf,
                  float* __restrict__ Mp, float* __restrict__ Lp) {
  int l = threadIdx.x, hf = l >> 4, lmn = l & 15;
  int qt = blockIdx.x, head = blockIdx.y, n = blockIdx.z;
  float runM[8], runS[8];
#pragma unroll
  for (int r = 0; r < 8; ++r) { runM[r] = -INFINITY; runS[r] = 0.f; }
  v2f av[8];
  const float* qp = qbuf + (size_t)(n * QP + qt * 16 + lmn) * DIMC + head * DH_C + 2 * hf;
#pragma unroll
  for (int s = 0; s < 8; ++s) av[s] = *(const v2f*)(qp + 4 * s);
  for (int t = 0; t < KK / 16; ++t) {
    const float* kp = kbuf + (size_t)(n * KK + t * 16 + lmn) * DIMC + head * DH_C + 2 * hf;
    __builtin_prefetch((const void*)(kp + 16 * DIMC), 0, 1);  // next K tile
    v2f bv2[8];
#pragma unroll
    for (int s = 0; s < 8; ++s) bv2[s] = *(const v2f*)(kp + 4 * s);  // clause of 8 b64 loads
    v8f s8 = {0.f, 0.f, 0.f, 0.f, 0.f, 0.f, 0.f, 0.f};
#pragma unroll
    for (int s = 0; s < 8; ++s) s8 = wmma4(av[s], bv2[s], s8);
#pragma unroll
    for (int r = 0; r < 8; ++r) {
      float xv = s8[r] * ATT_SCALE;
      float rm = xv;
      rm = fmaxf(rm, __shfl_xor(rm, 1, 32));
      rm = fmaxf(rm, __shfl_xor(rm, 2, 32));
      rm = fmaxf(rm, __shfl_xor(rm, 4, 32));
      rm = fmaxf(rm, __shfl_xor(rm, 8, 32));
      float e = __expf(xv - rm);
      e += __shfl_xor(e, 1, 32);
      e += __shfl_xor(e, 2, 32);
      e += __shfl_xor(e, 4, 32);
      e += __shfl_xor(e, 8, 32);
      float nm = fmaxf(runM[r], rm);
      runS[r] = runS[r] * __expf(runM[r] - nm) + e * __expf(rm - nm);
      runM[r] = nm;
    }
  }
  if (lmn == 0) {
#pragma unroll
    for (int r = 0; r < 8; ++r) {
      int row = qt * 16 + r + 8 * hf;
      Mp[(size_t)(n * HEADS_C + head) * QP + row] = runM[r];
      Lp[(size_t)(n * HEADS_C + head) * QP + row] = runS[r];
    }
  }
}

// combine per-camera (M,L) partials into global row max / normalizer
__global__ void k_attn_combine1(const float* __restrict__ Mp, const float* __restrict__ Lp,
                                float* __restrict__ Mr, float* __restrict__ Lr) {
  int i = blockIdx.x * blockDim.x + threadIdx.x;
  if (i >= HEADS_C * QP) return;
  int head = i / QP, row = i % QP;
  float m = -INFINITY;
#pragma unroll
  for (int n = 0; n < NCAM; ++n)
    m = fmaxf(m, Mp[(size_t)(n * HEADS_C + head) * QP + row]);
  float ll = 0.f;
#pragma unroll
  for (int n = 0; n < NCAM; ++n)
    ll += Lp[(size_t)(n * HEADS_C + head) * QP + row] *
          __expf(Mp[(size_t)(n * HEADS_C + head) * QP + row] - m);
  Mr[(size_t)head * QP + row] = m;
  Lr[(size_t)head * QP + row] = ll;
}

// --------------- attention pass 2 (per camera): A_n = P_n @ V_n -------------
__global__ __launch_bounds__(32)
void k_attn_pass2(const float* __restrict__ qbuf, const float* __restrict__ kbuf,
                  const float* __restrict__ vbuf,
                  const float* __restrict__ Mrow, const float* __restrict__ Lrow,
                  float* __restrict__ apart) {
  __shared__ float lds[16 * 17];
  int l = threadIdx.x, hf = l >> 4, lmn = l & 15;
  int qt = blockIdx.x, head = blockIdx.y, n = blockIdx.z;
  float mr[8], il[8];
#pragma unroll
  for (int r = 0; r < 8; ++r) {
    int row = qt * 16 + r + 8 * hf;
    mr[r] = Mrow[(size_t)head * QP + row];
    il[r] = 1.f / Lrow[(size_t)head * QP + row];
  }
  v8f acc0 = {0.f, 0.f, 0.f, 0.f, 0.f, 0.f, 0.f, 0.f};
  v8f acc1 = {0.f, 0.f, 0.f, 0.f, 0.f, 0.f, 0.f, 0.f};
  v2f av[8];
  const float* qp = qbuf + (size_t)(n * QP + qt * 16 + lmn) * DIMC + head * DH_C + 2 * hf;
#pragma unroll
  for (int s = 0; s < 8; ++s) av[s] = *(const v2f*)(qp + 4 * s);
  for (int t = 0; t < KK / 16; ++t) {
    int kbase = n * KK + t * 16;
    const float* kp = kbuf + (size_t)(kbase + lmn) * DIMC + head * DH_C + 2 * hf;
    __builtin_prefetch((const void*)(kp + 16 * DIMC), 0, 1);  // next K tile
    v2f bv2[8];
#pragma unroll
    for (int s = 0; s < 8; ++s) bv2[s] = *(const v2f*)(kp + 4 * s);
    v8f s8 = {0.f, 0.f, 0.f, 0.f, 0.f, 0.f, 0.f, 0.f};
#pragma unroll
    for (int s = 0; s < 8; ++s) s8 = wmma4(av[s], bv2[s], s8);
    // P = exp(S*scale - M) / L  -> LDS (row = query, col = key), stride 17
#pragma unroll
    for (int r = 0; r < 8; ++r)
      lds[(r + 8 * hf) * 17 + lmn] = __expf(s8[r] * ATT_SCALE - mr[r]) * il[r];
    __syncthreads();
    // preload P A-fragments (LDS) and V B-fragments (global), then WMMA chain
    float pax[4], pay[4], b0x[4], b0y[4], b1x[4], b1y[4];
#pragma unroll
    for (int j = 0; j < 4; ++j) {
      int kk = 4 * j;
      pax[j] = lds[lmn * 17 + kk + 2 * hf];
      pay[j] = lds[lmn * 17 + kk + 2 * hf + 1];
      const float* vp0 = vbuf + (size_t)(kbase + kk + 2 * hf) * DIMC + head * DH_C;
      const float* vp1 = vp0 + DIMC;
      b0x[j] = vp0[lmn];      b0y[j] = vp1[lmn];
      b1x[j] = vp0[16 + lmn]; b1y[j] = vp1[16 + lmn];
    }
#pragma unroll
    for (int j = 0; j < 4; ++j) {
      v2f a;  a.x = pax[j];  a.y = pay[j];
      v2f b0; b0.x = b0x[j]; b0.y = b0y[j];
      acc0 = wmma4(a, b0, acc0);
      v2f b1; b1.x = b1x[j]; b1.y = b1y[j];
      acc1 = wmma4(a, b1, acc1);
    }
    __syncthreads();
  }
  float* ap = apart + (size_t)n * QP * DIMC;
#pragma unroll
  for (int r = 0; r < 8; ++r) {
    int qrow = qt * 16 + r + 8 * hf;
    ap[(size_t)qrow * DIMC + head * DH_C + lmn] = acc0[r];
    ap[(size_t)qrow * DIMC + head * DH_C + 16 + lmn] = acc1[r];
  }
}

// sum per-camera partial attention outputs in fixed order (deterministic)
__global__ void k_attn_combine2(const float* __restrict__ apart, float* __restrict__ abuf) {
  int i = blockIdx.x * blockDim.x + threadIdx.x;
  if (i >= QP * DIMC) return;
  float s = 0.f;
#pragma unroll
  for (int n = 0; n < NCAM; ++n) s += apart[(size_t)n * QP * DIMC + i];
  abuf[i] = s;
}

// ---------------------------------------------------------------------------
extern "C" void kernel_launch(void* const* d_in, const int* in_sizes, int n_in,
                              void* d_out, int out_size, void* d_ws, size_t ws_size,
                              hipStream_t stream) {
  (void)in_sizes; (void)n_in; (void)out_size; (void)ws_size;
  const float* x           = (const float*)d_in[0];
  const float* feature     = (const float*)d_in[1];
  const float* I_inv       = (const float*)d_in[2];
  const float* E_inv       = (const float*)d_in[3];
  const float* image_plane = (const float*)d_in[4];
  const float* world       = (const float*)d_in[5];
  // params flattened in sorted-key (JAX pytree) order
  const float* b_bev  = (const float*)d_in[6];
  const float* bk     = (const float*)d_in[7];
  const float* bq     = (const float*)d_in[8];
  const float* bv     = (const float*)d_in[9];
  const float* fl_b   = (const float*)d_in[10];
  const float* fl_g   = (const float*)d_in[11];
  const float* fl_m   = (const float*)d_in[12];
  const float* fl_v   = (const float*)d_in[13];
  const float* fl_w   = (const float*)d_in[14];
  const float* fp_b   = (const float*)d_in[15];
  const float* fp_g   = (const float*)d_in[16];
  const float* fp_m   = (const float*)d_in[17];
  const float* fp_v   = (const float*)d_in[18];
  const float* fp_w   = (const float*)d_in[19];
  const float* k_ln_b = (const float*)d_in[20];
  const float* k_ln_g = (const float*)d_in[21];
  const float* mlp_b1 = (const float*)d_in[22];
  const float* mlp_b2 = (const float*)d_in[23];
  const float* mlp_w1 = (const float*)d_in[24];
  const float* mlp_w2 = (const float*)d_in[25];
  const float* post_b = (const float*)d_in[26];
  const float* post_g = (const float*)d_in[27];
  const float* pre_b  = (const float*)d_in[28];
  const float* pre_g  = (const float*)d_in[29];
  const float* proj_b = (const float*)d_in[30];
  const float* proj_w = (const float*)d_in[31];
  const float* q_ln_b = (const float*)d_in[32];
  const float* q_ln_g = (const float*)d_in[33];
  const float* v_ln_b = (const float*)d_in[34];
  const float* v_ln_g = (const float*)d_in[35];
  const float* w_bev  = (const float*)d_in[36];
  const float* w_cam  = (const float*)d_in[37];
  const float* w_img  = (const float*)d_in[38];
  const float* wk     = (const float*)d_in[39];
  const float* wq     = (const float*)d_in[40];
  const float* wv     = (const float*)d_in[41];

  float* ws = (float*)d_ws;
  size_t off = 0;
  auto take = [&](size_t nelem) { float* p = ws + off; off += nelem; return p; };
  const size_t NK = (size_t)NCAM * KK * DIMC;       // 1,290,240
  const size_t NQ = (size_t)NCAM * QP * DIMC;       // 1,929,216
  float* c_embed = take((size_t)NCAM * DIMC);
  float* actp_t  = take(NK);
  float* actl_t  = take(NK);
  float* keyin   = take(NK);
  float* valin   = take(NK);
  float* queryin = take(NQ);
  float* qln     = take(NQ);
  float* kln     = take(NK);
  float* vln     = take(NK);
  float* qbuf    = take(NQ);
  float* kbuf    = take(NK);
  float* vbuf    = take(NK);
  float* Mp      = take((size_t)NCAM * HEADS_C * QP);
  float* Lp      = take((size_t)NCAM * HEADS_C * QP);
  float* Mr      = take((size_t)HEADS_C * QP);
  float* Lr      = take((size_t)HEADS_C * QP);
  float* apart   = take((size_t)NCAM * QP * DIMC);
  float* abuf    = take((size_t)QP * DIMC);
  float* projout = take((size_t)QP * DIMC);
  float* z1      = take((size_t)QP * DIMC);
  float* hbuf    = take((size_t)QP * 2 * DIMC);
  float* z2      = take((size_t)QP * DIMC);

  // 1. camera embeddings; 2. image-plane embedding -> keyin; 3. query init
  k_cembed<<<(NCAM * DIMC + 63) / 64, 64, 0, stream>>>(E_inv, w_cam, c_embed);
  k_imgembed<<<(NCAM * KK + 255) / 256, 256, 0, stream>>>(I_inv, E_inv, image_plane, w_img,
                                                          c_embed, keyin);
  k_queryin<<<(NCAM * QQ + 255) / 256, 256, 0, stream>>>(world, w_bev, b_bev, c_embed, x, queryin);
  // 4. BN+ReLU activations (transposed) for both convs
  k_bnrelu<<<(int)((NK + 255) / 256), 256, 0, stream>>>(feature, fp_g, fp_b, fp_m, fp_v,
                                                        fl_g, fl_b, fl_m, fl_v, actp_t, actl_t);
  // 5/6. 1x1 convs: keyin += act_t @ fp_w^T ; valin = act_t @ fl_w^T
  k_gemm<<<dim3(NCAM * KK / 16, DIMC / 64), 32, 0, stream>>>(
      actp_t, DIMC, fp_w, keyin, nullptr, keyin, DIMC);
  k_gemm<<<dim3(NCAM * KK / 16, DIMC / 64), 32, 0, stream>>>(
      actl_t, DIMC, fl_w, nullptr, nullptr, valin, DIMC);
  // 7-9. LayerNorms
  int rq = NCAM * QP, rk = NCAM * KK;
  k_ln<<<(rq * 32 + 255) / 256, 256, 0, stream>>>(queryin, nullptr, 0, 0, q_ln_g, q_ln_b,
                                                  qln, (long)DIMC, 1L, rq);
  k_ln<<<(rk * 32 + 255) / 256, 256, 0, stream>>>(keyin, nullptr, 0, 0, k_ln_g, k_ln_b,
                                                  kln, (long)DIMC, 1L, rk);
  k_ln<<<(rk * 32 + 255) / 256, 256, 0, stream>>>(valin, nullptr, 0, 0, v_ln_g, v_ln_b,
                                                  vln, (long)DIMC, 1L, rk);
  // 10-12. Q/K/V projections
  k_gemm<<<dim3(rq / 16, DIMC / 64), 32, 0, stream>>>(qln, DIMC, wq, nullptr, bq, qbuf, DIMC);
  k_gemm<<<dim3(rk / 16, DIMC / 64), 32, 0, stream>>>(kln, DIMC, wk, nullptr, bk, kbuf, DIMC);
  k_gemm<<<dim3(rk / 16, DIMC / 64), 32, 0, stream>>>(vln, DIMC, wv, nullptr, bv, vbuf, DIMC);
  // 13/14. two-pass flash attention, split per camera + deterministic combines
  k_attn_pass1<<<dim3(QP / 16, HEADS_C, NCAM), 32, 0, stream>>>(qbuf, kbuf, Mp, Lp);
  k_attn_combine1<<<(HEADS_C * QP + 255) / 256, 256, 0, stream>>>(Mp, Lp, Mr, Lr);
  k_attn_pass2<<<dim3(QP / 16, HEADS_C, NCAM), 32, 0, stream>>>(qbuf, kbuf, vbuf, Mr, Lr, apart);
  k_attn_combine2<<<(QP * DIMC + 255) / 256, 256, 0, stream>>>(apart, abuf);
  // 15. output projection
  k_gemm<<<dim3(QP / 16, DIMC / 64), 32, 0, stream>>>(abuf, DIMC, proj_w, nullptr, proj_b,
                                                      projout, DIMC);
  // 16. z1 = LN(proj + x^T)
  k_ln<<<(QQ * 32 + 255) / 256, 256, 0, stream>>>(projout, x, 2, QQ, pre_g, pre_b,
                                                  z1, (long)DIMC, 1L, QQ);
  // 17/18. MLP hidden + exact GELU
  k_gemm<<<dim3(QP / 16, (2 * DIMC) / 64), 32, 0, stream>>>(z1, DIMC, mlp_w1, nullptr, mlp_b1,
                                                            hbuf, 2 * DIMC);
  k_gelu<<<(QP * 2 * DIMC + 255) / 256, 256, 0, stream>>>(hbuf, QP * 2 * DIMC);
  // 19. z2 = z1 + hbuf @ mlp_w2^T + b2
  k_gemm<<<dim3(QP / 16, DIMC / 64), 32, 0, stream>>>(hbuf, 2 * DIMC, mlp_w2, z1, mlp_b2,
                                                      z2, DIMC);
  // 20. final LN, transposed store into d_out (B, D, H, W)
  k_ln<<<(QQ * 32 + 255) / 256, 256, 0, stream>>>(z2, nullptr, 0, 0, post_g, post_b,
                                                  (float*)d_out, 1L, (long)QQ, QQ);
}